// HGTModel_63453846831248
// MI455X (gfx1250) — compile-verified
//
#include <hip/hip_runtime.h>
#include <math.h>

// ---------------------------------------------------------------------------
// HGT (heterogeneous graph transformer), 2 layers, fp32, for MI455X (gfx1250).
//
// Dense compute -> V_WMMA_F32_16X16X4_F32 (wave32 WMMA, fp32 in/out, numerics
// match the fp32 reference). Edge transforms re-associated into per-node
// batched GEMMs:
//   score = k[src] . (A @ q[dst])  -> precompute qt = q @ A^T   (dense GEMM)
//   msg   = v[src] @ M             -> precompute vm = v @ M     (dense GEMM)
// Edge passes are L2-resident gathers + float atomics (node tables are ~51MB,
// resident in the 192MB L2).
//
// GEMM kernel contract (all call sites satisfy it): M % 16 == 0, Kd % 4 == 0,
// Nd % (NT*16) == 0 for the chosen NT specialization. The only control flow
// left is the wave-uniform tail skip and the k-loop backedge, so EXEC is
// statically all-ones at every WMMA (ISA requirement).
//
// Workspace requirement: ~320 MB (layout in kernel_launch).
// ---------------------------------------------------------------------------

typedef __attribute__((ext_vector_type(2))) float v2f;
typedef __attribute__((ext_vector_type(8))) float v8f;

#define WAVES_PER_BLOCK 8

__device__ __forceinline__ v8f wmma_f32(v2f a, v2f b, v8f c)
{
    return __builtin_amdgcn_wmma_f32_16x16x4_f32(
        false, a, false, b, (short)0, c, false, false);
}

// Generic fp32 WMMA GEMM, specialized on NT (16-col subtiles per wave) and RELU:
//   C[z][m][n] = sum_k A[z][m][k] * B[z][k][n]
// A: element (m,k) at A + z*strideA + m*lda + k            (row-major)
// B: element (k,n) at B + z*strideB + k*sbk + n*sbn        (strides -> B or B^T)
// C: element (m,n) at C + z*strideC + m*ldc + n            (row-major)
// One wave computes a 16 x (16*NT) tile; k stepped by 4 per WMMA; rotated
// software pipeline (loads of step k+4 issued before WMMAs of step k).
template<int NT, int RELU>
__global__ void __launch_bounds__(256)
gemm_f32_wmma(const float* __restrict__ A, const float* __restrict__ B,
              float* __restrict__ C, int M, int Kd,
              int lda, int ldc, long sbk, long sbn,
              long strideA, long strideB, long strideC)
{
    const int wave = threadIdx.x >> 5;
    const int m0   = (blockIdx.y * WAVES_PER_BLOCK + wave) * 16;
    if (m0 >= M) return;                      // wave-uniform tail skip

    A += (long)blockIdx.z * strideA;
    B += (long)blockIdx.z * strideB;
    C += (long)blockIdx.z * strideC;

    const int lane  = threadIdx.x & 31;
    const int n0    = blockIdx.x * (NT * 16);
    const int laneM = lane & 15;
    const int half  = lane >> 4;              // 0 or 1
    const int koff  = half * 2;               // this lane holds K = {koff, koff+1}

    // A fragment base: row m0+laneM, k offset koff (contiguous pair -> b64 load)
    const float* arow = A + (long)(m0 + laneM) * lda + koff;
    // B fragment base: col n0+laneM, k offset koff
    const float* bcol = B + (long)(n0 + laneM) * sbn + (long)koff * sbk;
    const long   bt   = 16 * sbn;             // subtile column stride

    v8f acc[NT];
#pragma unroll
    for (int t = 0; t < NT; ++t) acc[t] = v8f{};

    // ---- prologue: fragments for k = 0 ----
    v2f ca, cb[NT];
    ca.x = arow[0];
    ca.y = arow[1];
#pragma unroll
    for (int t = 0; t < NT; ++t) {
        cb[t].x = bcol[t * bt];
        cb[t].y = bcol[t * bt + sbk];
    }

    // ---- steady state: load k, multiply k-4 ----
    for (int k = 4; k < Kd; k += 4) {
        v2f na, nb[NT];
        na.x = arow[k];
        na.y = arow[k + 1];
        const float* p = bcol + (long)k * sbk;
#pragma unroll
        for (int t = 0; t < NT; ++t) {
            nb[t].x = p[t * bt];
            nb[t].y = p[t * bt + sbk];
        }
#pragma unroll
        for (int t = 0; t < NT; ++t) acc[t] = wmma_f32(ca, cb[t], acc[t]);
        ca = na;
#pragma unroll
        for (int t = 0; t < NT; ++t) cb[t] = nb[t];
    }

    // ---- epilogue: multiply last k step ----
#pragma unroll
    for (int t = 0; t < NT; ++t) acc[t] = wmma_f32(ca, cb[t], acc[t]);

    // ---- store: VGPR g -> row m0 + 8*half + g, col n0 + 16*t + laneM ----
#pragma unroll
    for (int g = 0; g < 8; ++g) {
        const int rr = m0 + half * 8 + g;
        float* crow = C + (long)rr * ldc + n0 + laneM;
#pragma unroll
        for (int t = 0; t < NT; ++t) {
            float v = acc[t][g];
            if (RELU) v = fmaxf(v, 0.f);
            crow[t * 16] = v;
        }
    }
}

// Host-side dispatch: pick NT from Nd (all Nd are multiples of 16; NT in {1,2,4}).
static void gemm_launch(const float* A, const float* B, float* C,
                        int M, int Kd, int Nd, int lda, int ldc,
                        long sbk, long sbn, long sA, long sB, long sC,
                        int nbatch, int relu, hipStream_t stream)
{
    dim3 blk(256);
    int nt = Nd / 16;
    if (nt > 4) nt = 4;
    const unsigned gy = (unsigned)(((M + 15) / 16 + WAVES_PER_BLOCK - 1) / WAVES_PER_BLOCK);
    const unsigned gx = (unsigned)(Nd / (nt * 16));
    dim3 grid(gx, gy, (unsigned)nbatch);
#define GO(NTC, RLC)                                                        \
    gemm_f32_wmma<NTC, RLC><<<grid, blk, 0, stream>>>(                      \
        A, B, C, M, Kd, lda, ldc, sbk, sbn, sA, sB, sC)
    if (relu) {
        switch (nt) { case 1: GO(1, 1); break; case 2: GO(2, 1); break; default: GO(4, 1); break; }
    } else {
        switch (nt) { case 1: GO(1, 0); break; case 2: GO(2, 0); break; default: GO(4, 0); break; }
    }
#undef GO
}

// ---------------------------------------------------------------------------

__global__ void __launch_bounds__(256)
fill_f32(float* __restrict__ p, long n, float v)
{
    long i = (long)blockIdx.x * blockDim.x + threadIdx.x;
    if (i < n) p[i] = v;
}

__device__ __forceinline__ void atomicMaxF32(float* addr, float val)
{
    // IEEE trick: signed-int max for val>=0, unsigned-int min for val<0.
    // Works with buffer initialized to -inf (0xFF800000).
    if (val >= 0.f)
        atomicMax((int*)addr, __float_as_int(val));
    else
        atomicMin((unsigned int*)addr, __float_as_uint(val));
}

// Pass 1 (per relation): score = prior/sqrt(dh) * dot(k[src], qt[dst]);
// store score, atomic-max into per-(dst,head) running max.
__global__ void __launch_bounds__(256)
edge_score(const float* __restrict__ Kf, const float* __restrict__ QT,
           const int* __restrict__ ei, const float* __restrict__ prior,
           float* __restrict__ score, float* __restrict__ smax,
           int r, int E, int H, int dh, int d, float scale)
{
    long tid = (long)blockIdx.x * blockDim.x + threadIdx.x;
    if (tid >= (long)E * H) return;
    const int e = (int)(tid / H);
    const int h = (int)(tid % H);
    const int src = ei[((long)r * 2 + 0) * E + e];
    const int dst = ei[((long)r * 2 + 1) * E + e];
    const float4* kp = (const float4*)(Kf + (long)src * d + h * dh);
    const float4* qp = (const float4*)(QT + (long)dst * d + h * dh);
    float s = 0.f;
    for (int i = 0; i < dh / 4; ++i) {
        float4 a = kp[i], b = qp[i];
        s = fmaf(a.x, b.x, fmaf(a.y, b.y, fmaf(a.z, b.z, fmaf(a.w, b.w, s))));
    }
    s *= prior[r * H + h] * scale;
    score[((long)r * E + e) * (long)H + h] = s;
    atomicMaxF32(smax + (long)dst * H + h, s);
}

// smax = where(isfinite(smax), smax, 0)  (nodes with no in-edges stay -inf)
__global__ void __launch_bounds__(256)
smax_fix(float* __restrict__ smax, long n)
{
    long i = (long)blockIdx.x * blockDim.x + threadIdx.x;
    if (i < n) {
        float v = smax[i];
        if (!__builtin_isfinite(v)) smax[i] = 0.f;
    }
}

// Pass 2 (all relations): ex = exp(score - smax[dst]); denom[dst] += ex.
__global__ void __launch_bounds__(256)
edge_exp(const int* __restrict__ ei, const float* __restrict__ smax,
         float* __restrict__ score, float* __restrict__ denom,
         int R, int E, int H)
{
    long tid = (long)blockIdx.x * blockDim.x + threadIdx.x;
    if (tid >= (long)R * E * H) return;
    const long ge = tid / H;
    const int  h  = (int)(tid % H);
    const int  r  = (int)(ge / E);
    const int  e  = (int)(ge % E);
    const int dst = ei[((long)r * 2 + 1) * E + e];
    const float ex = expf(score[tid] - smax[(long)dst * H + h]);
    score[tid] = ex;
    atomicAdd(denom + (long)dst * H + h, ex);
}

// Pass 3 (per relation): aggr[dst] += vm[src] * ex/(denom[dst]+1e-9)
__global__ void __launch_bounds__(256)
edge_aggr(const float* __restrict__ VM, const int* __restrict__ ei,
          const float* __restrict__ score, const float* __restrict__ denom,
          float* __restrict__ aggr, int r, int E, int H, int dh, int d)
{
    long tid = (long)blockIdx.x * blockDim.x + threadIdx.x;
    if (tid >= (long)E * H) return;
    const int e = (int)(tid / H);
    const int h = (int)(tid % H);
    const int src = ei[((long)r * 2 + 0) * E + e];
    const int dst = ei[((long)r * 2 + 1) * E + e];
    const float ex    = score[((long)r * E + e) * (long)H + h];
    const float alpha = ex / (denom[(long)dst * H + h] + 1e-9f);
    const float4* vp = (const float4*)(VM + (long)src * d + h * dh);
    float* op = aggr + (long)dst * d + h * dh;
    for (int i = 0; i < dh / 4; ++i) {
        float4 m = vp[i];
        atomicAdd(op + 4 * i + 0, m.x * alpha);
        atomicAdd(op + 4 * i + 1, m.y * alpha);
        atomicAdd(op + 4 * i + 2, m.z * alpha);
        atomicAdd(op + 4 * i + 3, m.w * alpha);
    }
}

// ---------------------------------------------------------------------------

static void run_layer(const float* X, int din, int dout,
                      const float* Wk, const float* Wq, const float* Wv,
                      const float* Aw, const float* Mw,
                      const float* prior, const float* Wo,
                      const int* ei,
                      float* Kb, float* Qb, float* Vb, float* Tb,
                      float* scoreb, float* smaxb, float* denomb, float* aggrb,
                      float* outb, int relu, hipStream_t stream)
{
    const int Nn = 100000, R = 3, E = 200000, H = 4;
    const int dh = dout / H;
    dim3 blk(256);

    // K/Q/V node projections: (N x din) @ (din x dout)
    gemm_launch(X, Wk, Kb, Nn, din, dout, din, dout, dout, 1, 0, 0, 0, 1, 0, stream);
    gemm_launch(X, Wq, Qb, Nn, din, dout, din, dout, dout, 1, 0, 0, 0, 1, 0, stream);
    gemm_launch(X, Wv, Vb, Nn, din, dout, din, dout, dout, 1, 0, 0, 0, 1, 0, stream);

    const long nh = (long)Nn * H;
    const long nd = (long)Nn * dout;
    fill_f32<<<(unsigned)((nh + 255) / 256), blk, 0, stream>>>(smaxb, nh, -__builtin_inff());
    fill_f32<<<(unsigned)((nh + 255) / 256), blk, 0, stream>>>(denomb, nh, 0.f);
    fill_f32<<<(unsigned)((nd + 255) / 256), blk, 0, stream>>>(aggrb, nd, 0.f);

    const float scale = 1.0f / sqrtf((float)dh);
    const long  eh    = (long)E * H;

    // Scores + running segment max
    for (int r = 0; r < R; ++r) {
        // qt[n, h*dh + d'] = sum_f Q[n, h*dh+f] * A[r,h,d',f]
        // B(k=f, n=d') = A[d'*dh + f]  ->  sbk=1, sbn=dh ; batched over heads.
        gemm_launch(Qb, Aw + (long)r * H * dh * dh, Tb, Nn, dh, dh, dout, dout,
                    /*sbk=*/1, /*sbn=*/dh, /*sA=*/dh, /*sB=*/(long)dh * dh,
                    /*sC=*/dh, H, 0, stream);
        edge_score<<<(unsigned)((eh + 255) / 256), blk, 0, stream>>>(
            Kb, Tb, ei, prior, scoreb, smaxb, r, E, H, dh, dout, scale);
    }
    smax_fix<<<(unsigned)((nh + 255) / 256), blk, 0, stream>>>(smaxb, nh);

    const long reh = (long)R * E * H;
    edge_exp<<<(unsigned)((reh + 255) / 256), blk, 0, stream>>>(
        ei, smaxb, scoreb, denomb, R, E, H);

    // Messages + normalized scatter-add
    for (int r = 0; r < R; ++r) {
        // vm[n, h*dh + f] = sum_d V[n, h*dh+d] * M[r,h,d,f]
        // B(k=d, n=f) = M[d*dh + f]  ->  sbk=dh, sbn=1 ; batched over heads.
        gemm_launch(Vb, Mw + (long)r * H * dh * dh, Tb, Nn, dh, dh, dout, dout,
                    /*sbk=*/dh, /*sbn=*/1, dh, (long)dh * dh, dh, H, 0, stream);
        edge_aggr<<<(unsigned)((eh + 255) / 256), blk, 0, stream>>>(
            Tb, ei, scoreb, denomb, aggrb, r, E, H, dh, dout);
    }

    // Output projection (+ReLU for layer 1)
    gemm_launch(aggrb, Wo, outb, Nn, dout, dout, dout, dout, dout, 1, 0, 0, 0, 1,
                relu, stream);
}

extern "C" void kernel_launch(void* const* d_in, const int* in_sizes, int n_in,
                              void* d_out, int out_size, void* d_ws, size_t ws_size,
                              hipStream_t stream)
{
    (void)in_sizes; (void)n_in; (void)out_size; (void)ws_size;

    const float* x      = (const float*)d_in[0];
    const int*   ei     = (const int*)d_in[1];
    const float* Wk1    = (const float*)d_in[2];
    const float* Wq1    = (const float*)d_in[3];
    const float* Wv1    = (const float*)d_in[4];
    const float* A1     = (const float*)d_in[5];
    const float* M1     = (const float*)d_in[6];
    const float* prior1 = (const float*)d_in[7];
    const float* Wo1    = (const float*)d_in[8];
    const float* Wk2    = (const float*)d_in[9];
    const float* Wq2    = (const float*)d_in[10];
    const float* Wv2    = (const float*)d_in[11];
    const float* A2     = (const float*)d_in[12];
    const float* M2     = (const float*)d_in[13];
    const float* prior2 = (const float*)d_in[14];
    const float* Wo2    = (const float*)d_in[15];

    const long Nn = 100000;
    const int  HID = 128, OUT = 64, H = 4, R = 3, E = 200000;
    const long NH = Nn * (long)HID;

    // Workspace layout (floats); total = 6*NH + R*E*H + 2*N*H
    //   = 6*12.8M + 2.4M + 0.8M = 80.0M floats = 320 MB.
    float* ws     = (float*)d_ws;
    float* Kb     = ws;                               // N*HID (layer2 uses N*OUT subset)
    float* Qb     = Kb + NH;                          // N*HID
    float* Vb     = Qb + NH;                          // N*HID
    float* Tb     = Vb + NH;                          // N*HID  (qt / vm temp)
    float* scoreb = Tb + NH;                          // R*E*H  (scores, then exp)
    float* smaxb  = scoreb + (long)R * E * H;         // N*H
    float* denomb = smaxb + Nn * H;                   // N*H
    float* aggrb  = denomb + Nn * H;                  // N*HID
    float* h1     = aggrb + NH;                       // N*HID  (layer-1 activations)

    // Layer 1: 128 -> 128, ReLU
    run_layer(x, 128, HID, Wk1, Wq1, Wv1, A1, M1, prior1, Wo1, ei,
              Kb, Qb, Vb, Tb, scoreb, smaxb, denomb, aggrb, h1, /*relu=*/1, stream);

    // Layer 2: 128 -> 64, writes final output
    run_layer(h1, HID, OUT, Wk2, Wq2, Wv2, A2, M2, prior2, Wo2, ei,
              Kb, Qb, Vb, Tb, scoreb, smaxb, denomb, aggrb,
              (float*)d_out, /*relu=*/0, stream);
}